// MoNet_8100308320979
// MI455X (gfx1250) — compile-verified
//
#include <hip/hip_runtime.h>

typedef __attribute__((ext_vector_type(2))) float v2f;
typedef __attribute__((ext_vector_type(8))) float v8f;

#define FDIM 64     // in/out features
#define KF   192    // K * F
#define CTILES 12   // KF / 16 column tiles
#define NKERN 3     // Gaussian kernels
#define NLAYER 3

// ---------------------------------------------------------------------------
// Kernel 1: hp = h @ W  (n x 64) @ (64 x 192), fp32 via V_WMMA_F32_16X16X4_F32.
// One wave per 16-row M-tile computing ALL 12 column tiles (16x192 of output):
// the A fragment for each K-step is loaded once and reused by 12 WMMAs,
// eliminating the 12x redundant A traffic of a tile-per-wave layout.
// K=64 -> 16 steps of K=4 -> 192 v_wmma per wave.
//
// fp32 WMMA VGPR layouts (ISA 7.12.2):
//   A 16x4 : lanes 0-15 hold M=0..15 {K=k0,k0+1}, lanes 16-31 hold {K=k0+2,k0+3}
//   B 4x16 : VGPR0 = rows K=k0 (lanes 0-15) / K=k0+2 (lanes 16-31), VGPR1 = +1
//   C/D    : VGPR r = row M=r (lanes 0-15, N=lane) / M=r+8 (lanes 16-31)
// ---------------------------------------------------------------------------
__global__ __launch_bounds__(32)
void gmm_gemm_wmma(const float* __restrict__ A,   // [n, 64]
                   const float* __restrict__ W,   // [64, 192]
                   float* __restrict__ HP,        // [n, 192]
                   int n)
{
    const int lane = threadIdx.x & 31;
    const int half = lane >> 4;       // 0: lanes 0-15, 1: lanes 16-31
    const int l15  = lane & 15;
    const int m0   = blockIdx.x * 16;

    int row = m0 + l15;
    if (row >= n) row = n - 1;        // clamp loads; stores are masked below

    v8f acc[CTILES];
#pragma unroll
    for (int c = 0; c < CTILES; ++c) acc[c] = (v8f){};

#pragma unroll
    for (int k0 = 0; k0 < FDIM; k0 += 4) {
        const int ka = k0 + half * 2;
        // A fragment: 2 consecutive fp32 along K for this lane's row (reused 12x)
        const v2f a = *(const v2f*)(A + (size_t)row * FDIM + ka);
        const float* w0 = W + (size_t)ka * KF + l15;
        const float* w1 = w0 + KF;
#pragma unroll
        for (int c = 0; c < CTILES; ++c) {
            v2f b;
            b.x = w0[c * 16];
            b.y = w1[c * 16];
            acc[c] = __builtin_amdgcn_wmma_f32_16x16x4_f32(
                /*neg_a=*/false, a, /*neg_b=*/false, b,
                /*c_mod=*/(short)0, acc[c], /*reuse_a=*/false, /*reuse_b=*/false);
        }
    }

#pragma unroll
    for (int r = 0; r < 8; ++r) {
        const int m = m0 + r + half * 8;
        if (m < n) {
            float* dst = HP + (size_t)m * KF + l15;
#pragma unroll
            for (int c = 0; c < CTILES; ++c) dst[c * 16] = acc[c][r];
        }
    }
}

// ---------------------------------------------------------------------------
// Kernel 2: per-edge Gaussian mixture weights + gather + segment-sum.
// One wave32 per destination node: lane j computes the weights for edge j
// (DEG==32 here; general chunked loop kept), then each edge's (w0,w1,w2,src)
// is broadcast via v_readlane (uniform index -> SGPRs, no LDS round-trip) and
// all 32 lanes gather hp[src] rows (2 features per lane as float2).
// hp (38.4 MB) is resident in MI455X's 192 MB L2, so this loop runs at L2 BW.
// ---------------------------------------------------------------------------
__global__ __launch_bounds__(256)
void gmm_aggregate(const float* __restrict__ HP,      // [n, 192]
                   const float* __restrict__ pseudo,  // [e, 2]
                   const int*   __restrict__ rowptr,  // [n+1]
                   const int*   __restrict__ colind,  // [e]
                   const float* __restrict__ projW,   // [2,2] this layer
                   const float* __restrict__ projB,   // [2]
                   const float* __restrict__ mu,      // [3,2]
                   const float* __restrict__ isig,    // [3,2]
                   float* __restrict__ out,           // [n, 64]
                   int n)
{
    const int wid  = threadIdx.x >> 5;
    const int lane = threadIdx.x & 31;
    const int node = blockIdx.x * 8 + wid;
    if (node >= n) return;

    // Uniform per-layer parameters (scalar loads)
    const float w00 = projW[0], w01 = projW[1];   // proj_W[in][dim]
    const float w10 = projW[2], w11 = projW[3];
    const float b0 = projB[0], b1 = projB[1];
    float mu0[NKERN], mu1[NKERN], s0[NKERN], s1[NKERN];
#pragma unroll
    for (int k = 0; k < NKERN; ++k) {
        mu0[k] = mu[2 * k];
        mu1[k] = mu[2 * k + 1];
        const float a = isig[2 * k], c = isig[2 * k + 1];
        s0[k] = a * a;
        s1[k] = c * c;
    }

    const int beg = rowptr[node];
    const int end = rowptr[node + 1];

    float accx = 0.f, accy = 0.f;
    for (int e0 = beg; e0 < end; e0 += 32) {
        const int e = e0 + lane;
        float wk[NKERN] = {0.f, 0.f, 0.f};
        int src = 0;
        if (e < end) {
            const float p0 = pseudo[2 * (size_t)e];
            const float p1 = pseudo[2 * (size_t)e + 1];
            const float u0 = tanhf(p0 * w00 + p1 * w10 + b0);
            const float u1 = tanhf(p0 * w01 + p1 * w11 + b1);
#pragma unroll
            for (int k = 0; k < NKERN; ++k) {
                const float d0 = u0 - mu0[k];
                const float d1 = u1 - mu1[k];
                wk[k] = __expf(-0.5f * (d0 * d0 * s0[k] + d1 * d1 * s1[k]));
            }
            src = colind[e];
        }
        const int cnt = min(32, end - e0);
        for (int j = 0; j < cnt; ++j) {
            // j is wave-uniform -> v_readlane into SGPRs (no ds_bpermute)
            const float g0 = __uint_as_float(
                __builtin_amdgcn_readlane(__float_as_uint(wk[0]), j));
            const float g1 = __uint_as_float(
                __builtin_amdgcn_readlane(__float_as_uint(wk[1]), j));
            const float g2 = __uint_as_float(
                __builtin_amdgcn_readlane(__float_as_uint(wk[2]), j));
            const int s = __builtin_amdgcn_readlane(src, j);
            const v2f* hpr = (const v2f*)(HP + (size_t)s * KF);
            const v2f r0 = hpr[lane];         // k=0 block, feats 2*lane..+1
            const v2f r1 = hpr[32 + lane];    // k=1 block
            const v2f r2 = hpr[64 + lane];    // k=2 block
            accx += g0 * r0.x + g1 * r1.x + g2 * r2.x;
            accy += g0 * r0.y + g1 * r1.y + g2 * r2.y;
        }
    }

    v2f res;
    res.x = accx;
    res.y = accy;
    *(v2f*)(out + (size_t)node * FDIM + 2 * lane) = res;
}

// ---------------------------------------------------------------------------
// Host-side launcher: 3 layers, each = GEMM (WMMA) + aggregate.
// Workspace: hp [n*192 f32] then h ping-pong buffer [n*64 f32] (~51.2 MB).
// Aggregation only reads hp, so h can be overwritten in place each layer.
// ---------------------------------------------------------------------------
extern "C" void kernel_launch(void* const* d_in, const int* in_sizes, int n_in,
                              void* d_out, int out_size, void* d_ws, size_t ws_size,
                              hipStream_t stream) {
    const float* feat   = (const float*)d_in[0];
    const float* pseudo = (const float*)d_in[1];
    const int*   rowptr = (const int*)d_in[2];
    const int*   colind = (const int*)d_in[3];
    const float* proj_W = (const float*)d_in[4];   // [L,2,2]
    const float* proj_b = (const float*)d_in[5];   // [L,2]
    const float* fc_W   = (const float*)d_in[6];   // [L,64,192]
    const float* mu     = (const float*)d_in[7];   // [L,3,2]
    const float* isig   = (const float*)d_in[8];   // [L,3,2]
    float* out = (float*)d_out;

    const int n = in_sizes[2] - 1;   // rowptr has n+1 entries

    float* hp   = (float*)d_ws;                 // [n,192]
    float* hbuf = hp + (size_t)n * KF;          // [n,64]

    dim3 gemm_grid((n + 15) / 16);
    dim3 agg_grid((n + 7) / 8);

    for (int l = 0; l < NLAYER; ++l) {
        const float* hin  = (l == 0) ? feat : hbuf;
        float*       hout = (l == NLAYER - 1) ? out : hbuf;
        gmm_gemm_wmma<<<gemm_grid, 32, 0, stream>>>(
            hin, fc_W + (size_t)l * FDIM * KF, hp, n);
        gmm_aggregate<<<agg_grid, 256, 0, stream>>>(
            hp, pseudo, rowptr, colind,
            proj_W + l * 4, proj_b + l * 2, mu + l * 6, isig + l * 6,
            hout, n);
    }
}